// EuclidFartherSubsample_17952963297847
// MI455X (gfx1250) — compile-verified
//
#include <hip/hip_runtime.h>
#include <hip/hip_bf16.h>
#include <stdint.h>

#define FPS_N   8192
#define FPS_M   4096
#define FPS_B   16
#define FPS_D   256
#define THREADS 1024
#define PPT     (FPS_N / THREADS)   // 8 points per thread

// ---------------------------------------------------------------------------
// Kernel 1: farthest-point sampling. One workgroup (32 waves) per batch.
// Coords live in LDS (96KB, DMA'd in via the Tensor Data Mover), the
// per-thread 8 points + 8 running distances live in registers.
// ---------------------------------------------------------------------------
__global__ __launch_bounds__(THREADS) void fps_kernel(
    const float* __restrict__ coords,   // (B, N, 3)
    const int*   __restrict__ init_idx, // (B,)
    int*         __restrict__ idx_out)  // (B, M)
{
    __shared__ float sc[FPS_N * 3];     // 96 KB coords tile
    __shared__ float swv[32];
    __shared__ int   swi[32];
    __shared__ int   sFar;

    const int tid  = threadIdx.x;
    const int lane = tid & 31;
    const int wid  = tid >> 5;
    const int b    = blockIdx.x;
    const float* cb = coords + (size_t)b * FPS_N * 3;

    // ---- Stage coords into LDS ------------------------------------------
#if defined(__gfx1250__) && __has_builtin(__builtin_amdgcn_tensor_load_to_lds)
    {
        // TDM ignores EXEC, so gate by a *scalar* wave id so only wave 0's
        // instruction stream ever reaches the tensor op.
        int wave_sgpr = __builtin_amdgcn_readfirstlane(tid >> 5);
        if (wave_sgpr == 0) {
            typedef unsigned int u32x4 __attribute__((ext_vector_type(4)));
            typedef int          i32x8 __attribute__((ext_vector_type(8)));
            typedef int          i32x4 __attribute__((ext_vector_type(4)));

            const uint32_t elems = FPS_N * 3;                 // 24576 dwords
            uint64_t gaddr = (uint64_t)(uintptr_t)cb;
            uint32_t laddr = (uint32_t)(uintptr_t)(void*)sc;  // generic addr low32 == LDS offset

            u32x4 g0;
            g0[0] = 1u;                                        // count=1, user descriptor
            g0[1] = laddr;                                     // lds_addr
            g0[2] = (uint32_t)(gaddr & 0xFFFFFFFFu);           // global_addr[31:0]
            g0[3] = (uint32_t)((gaddr >> 32) & 0x01FFFFFFu)    // global_addr[56:32]
                  | (2u << 30);                                // type = 2 ("image")

            i32x8 g1;
            g1[0] = (int)(2u << 16);                           // data_size = 4B
            g1[1] = (int)((elems & 0xFFFFu) << 16);            // tensor_dim0 lo16
            g1[2] = (int)(((elems >> 16) & 0xFFFFu)            // tensor_dim0 hi16
                  | (1u << 16));                               // tensor_dim1 = 1
            g1[3] = (int)((elems & 0xFFFFu) << 16);            // tile_dim0 = 24576
            g1[4] = 1;                                         // tile_dim1 = 1, tile_dim2 = 0
            g1[5] = (int)elems;                                // tensor_dim0_stride lo32
            g1[6] = 0;                                         // stride hi / dim1_stride lo
            g1[7] = 0;

            i32x4 gz4; gz4[0] = 0; gz4[1] = 0; gz4[2] = 0; gz4[3] = 0;
            i32x8 gz8;
            gz8[0] = 0; gz8[1] = 0; gz8[2] = 0; gz8[3] = 0;
            gz8[4] = 0; gz8[5] = 0; gz8[6] = 0; gz8[7] = 0;
            // 6-arg form (clang-23 / therock-10.0 headers)
            __builtin_amdgcn_tensor_load_to_lds(g0, g1, gz4, gz4, gz8, 0);
        }
#if __has_builtin(__builtin_amdgcn_s_wait_tensorcnt)
        __builtin_amdgcn_s_wait_tensorcnt(0);
#else
        asm volatile("s_wait_tensorcnt 0" ::: "memory");
#endif
    }
#else
    for (int i = tid; i < FPS_N * 3; i += THREADS) sc[i] = cb[i];
#endif
    if (tid == 0) sFar = init_idx[b];
    __syncthreads();

    // ---- Pull this thread's 8 points into registers ---------------------
    float px[PPT], py[PPT], pz[PPT], dst[PPT];
#pragma unroll
    for (int k = 0; k < PPT; ++k) {
        int i = tid + k * THREADS;
        px[k] = sc[i * 3 + 0];
        py[k] = sc[i * 3 + 1];
        pz[k] = sc[i * 3 + 2];
        dst[k] = 1e8f;                       // BIG, matches reference init
    }

    // ---- Sequential FPS loop --------------------------------------------
    for (int s = 0; s < FPS_M; ++s) {
        int far = sFar;                      // stable: last write fenced by barrier
        if (tid == 0) idx_out[(size_t)b * FPS_M + s] = far;   // record BEFORE update
        float cx = sc[far * 3 + 0];
        float cy = sc[far * 3 + 1];
        float cz = sc[far * 3 + 2];

        float bestv = -1.0f;
        int   besti = 0;
#pragma unroll
        for (int k = 0; k < PPT; ++k) {
            float dx = px[k] - cx, dy = py[k] - cy, dz = pz[k] - cz;
            float d  = dx * dx + dy * dy + dz * dz;
            float nd = fminf(dst[k], d);
            dst[k] = nd;
            int i = tid + k * THREADS;
            if (nd > bestv) { bestv = nd; besti = i; }   // i strictly increases w/ k
        }

        // wave32 argmax (tie -> smaller index, matching jnp.argmax)
#pragma unroll
        for (int off = 16; off > 0; off >>= 1) {
            float ov = __shfl_xor(bestv, off, 32);
            int   oi = __shfl_xor(besti, off, 32);
            if (ov > bestv || (ov == bestv && oi < besti)) { bestv = ov; besti = oi; }
        }
        if (lane == 0) { swv[wid] = bestv; swi[wid] = besti; }
        __syncthreads();

        if (wid == 0) {
            float v = swv[lane];
            int   i2 = swi[lane];
#pragma unroll
            for (int off = 16; off > 0; off >>= 1) {
                float ov = __shfl_xor(v, off, 32);
                int   oi = __shfl_xor(i2, off, 32);
                if (ov > v || (ov == v && oi < i2)) { v = ov; i2 = oi; }
            }
            if (lane == 0) sFar = i2;
        }
        __syncthreads();
    }
}

// ---------------------------------------------------------------------------
// Kernel 2: bandwidth-bound gather of coords/values/mask by sampled index.
// 4 rows per 256-thread block, 64 lanes x float4 per 256-float value row.
// ---------------------------------------------------------------------------
__global__ __launch_bounds__(256) void gather_kernel(
    const float* __restrict__ coords,   // (B, N, 3)
    const float* __restrict__ values,   // (B, N, 256)
    const float* __restrict__ maskin,   // (B, N)
    const int*   __restrict__ idx,      // (B, M) flat
    float* __restrict__ out_coords,     // (B, M, 3)
    float* __restrict__ out_values,     // (B, M, 256)
    float* __restrict__ out_mask)       // (B, M)
{
    const int tid  = threadIdx.x;
    const int row  = blockIdx.x * 4 + (tid >> 6);   // [0, B*M)
    const int lane = tid & 63;
    const int b    = row >> 12;                      // M = 4096
    const int id   = idx[row];
    const size_t src_row = (size_t)b * FPS_N + id;

    const float4* src = (const float4*)(values + src_row * FPS_D);
    float4*       dst = (float4*)(out_values + (size_t)row * FPS_D);
    dst[lane] = src[lane];

    if (lane == 0) {
        out_mask[row] = maskin[src_row];
        const float* c = coords + src_row * 3;
        float* oc = out_coords + (size_t)row * 3;
        oc[0] = c[0]; oc[1] = c[1]; oc[2] = c[2];
    }
}

// ---------------------------------------------------------------------------
extern "C" void kernel_launch(void* const* d_in, const int* in_sizes, int n_in,
                              void* d_out, int out_size, void* d_ws, size_t ws_size,
                              hipStream_t stream) {
    const float* coords  = (const float*)d_in[0];  // (16, 8192, 3)
    const float* values  = (const float*)d_in[1];  // (16, 8192, 256)
    const float* maskin  = (const float*)d_in[2];  // (16, 8192)
    const int*   init_ix = (const int*)d_in[3];    // (16,)

    float* out        = (float*)d_out;
    float* out_coords = out;                                      // 16*4096*3
    float* out_values = out + (size_t)FPS_B * FPS_M * 3;          // 16*4096*256
    float* out_mask   = out_values + (size_t)FPS_B * FPS_M * FPS_D;

    int* idx_ws = (int*)d_ws;                                     // 16*4096 ints

    fps_kernel<<<FPS_B, THREADS, 0, stream>>>(coords, init_ix, idx_ws);

    const int rows = FPS_B * FPS_M;                               // 65536
    gather_kernel<<<rows / 4, 256, 0, stream>>>(coords, values, maskin, idx_ws,
                                                out_coords, out_values, out_mask);
}